// SelfAttention_72000831750847
// MI455X (gfx1250) — compile-verified
//
#include <hip/hip_runtime.h>
#include <cstdint>

typedef uint32_t u32;
typedef uint64_t u64;
typedef __attribute__((ext_vector_type(2))) float v2f;
typedef __attribute__((ext_vector_type(8))) float v8f;
typedef __attribute__((ext_vector_type(4))) u32  v4u;
typedef __attribute__((ext_vector_type(4))) int  v4i;
typedef __attribute__((ext_vector_type(8))) int  v8i;

#define HD      64
#define PASTN   8192
#define TOTALK  8208          // PAST + T
#define CHUNK   128
#define NCHUNK  65            // 64 full chunks from past + 1 tail (16 new keys)
#define LDSTR   68            // 64 dwords + 4 pad dwords (16B-aligned rows, conflict-free frags)
#define PSTR    132           // P row stride (16 | (4r+c)%64 distinct)

// ---------------------------------------------------------------------------
// TDM: async 2D tile load (rows x 64 f32) Global -> LDS, with row padding
// 64 -> 68 dwords inserted by the TDM itself (pad_interval=64dw, pad_amount=4dw).
// tensor_dim1 = rows_valid clamps OOB rows to zero (used for the 16-row tail).
// ---------------------------------------------------------------------------
__device__ __forceinline__ void tdm_load_tile(u32 lds_byte_off, const float* gptr,
                                              u32 rows_valid) {
  union { v4u v; u32 a[4]; } g0;
  union { v8i v; u32 a[8]; } g1;
  u64 ga = (u64)(uintptr_t)gptr;
  g0.a[0] = 1u;                                   // count=1, user descriptor
  g0.a[1] = lds_byte_off;                         // lds_addr
  g0.a[2] = (u32)ga;                              // global_addr lo
  g0.a[3] = (u32)((ga >> 32) & 0x1FFFFFFu) | (2u << 30);  // addr hi | type=2
  g1.a[0] = (2u << 16)   // data_size = 4B
          | (1u << 20)   // pad_enable
          | (5u << 22)   // pad_interval: 64 dwords
          | (3u << 25);  // pad_amount: 4 dwords
  g1.a[1] = (64u & 0xFFFFu) << 16;                // tensor_dim0 = 64 (lo16)
  g1.a[2] = (rows_valid & 0xFFFFu) << 16;         // tensor_dim1 lo16
  g1.a[3] = (rows_valid >> 16) | (64u << 16);     // tensor_dim1 hi16 | tile_dim0 = 64
  g1.a[4] = (u32)CHUNK;                           // tile_dim1 = 128, tile_dim2 = 0
  g1.a[5] = 64u;                                  // tensor_dim0_stride = 64 elems
  g1.a[6] = 0u;
  g1.a[7] = 0u;
  v4i gz = {0, 0, 0, 0};
#if defined(__clang_major__) && (__clang_major__ >= 23)
  v8i z8 = {0, 0, 0, 0, 0, 0, 0, 0};
  __builtin_amdgcn_tensor_load_to_lds(g0.v, g1.v, gz, gz, z8, 0);
#else
  __builtin_amdgcn_tensor_load_to_lds(g0.v, g1.v, gz, gz, 0);
#endif
}

// ---------------------------------------------------------------------------
// GEMM: out = A[128x1024] @ W^T[1024x1024] + bias, f32 WMMA 16x16x4.
// out_mode 0: write [B,H,T,HD] head layout (q/k/v). out_mode 1: row-major.
// grid = 16 (64-col panel == one head), block = 128 (4 waves).
// ---------------------------------------------------------------------------
__global__ void __launch_bounds__(128)
proj_kernel(const float* __restrict__ A, const float* __restrict__ W,
            const float* __restrict__ bias, float* __restrict__ out,
            int out_mode, float scale) {
  __shared__ __align__(16) float As[128 * LDSTR];
  __shared__ __align__(16) float Ws[64 * LDSTR];

  const int tid   = threadIdx.x;
  const int lane  = tid & 31;
  const int wave  = tid >> 5;
  const int lhalf = lane >> 4;
  const int lmod  = lane & 15;
  const int panel = blockIdx.x;

  v8f acc[8];
#pragma unroll
  for (int m = 0; m < 8; ++m) acc[m] = {};

  for (int kb = 0; kb < 1024; kb += 64) {
#pragma unroll
    for (int i = 0; i < 16; ++i) {          // x chunk: 128 rows x 64
      int idx = tid + i * 128, row = idx >> 4, c4 = idx & 15;
      float4 t4 = *(const float4*)(A + row * 1024 + kb + c4 * 4);
      *(float4*)&As[row * LDSTR + c4 * 4] = t4;
    }
#pragma unroll
    for (int i = 0; i < 8; ++i) {           // W chunk: 64 rows x 64
      int idx = tid + i * 128, row = idx >> 4, c4 = idx & 15;
      float4 t4 = *(const float4*)(W + (size_t)(panel * 64 + row) * 1024 + kb + c4 * 4);
      *(float4*)&Ws[row * LDSTR + c4 * 4] = t4;
    }
    __syncthreads();
#pragma unroll 4
    for (int k = 0; k < 16; ++k) {
      v2f bfrag = *(const v2f*)&Ws[(wave * 16 + lmod) * LDSTR + 4 * k + 2 * lhalf];
#pragma unroll
      for (int m = 0; m < 8; ++m) {
        v2f afrag = *(const v2f*)&As[(m * 16 + lmod) * LDSTR + 4 * k + 2 * lhalf];
        acc[m] = __builtin_amdgcn_wmma_f32_16x16x4_f32(
            false, afrag, false, bfrag, (short)0, acc[m], false, false);
      }
    }
    __syncthreads();
  }

  const int gn = panel * 64 + wave * 16 + lmod;
  const float bv = bias[gn];
#pragma unroll
  for (int m = 0; m < 8; ++m) {
#pragma unroll
    for (int r = 0; r < 8; ++r) {
      float y = (acc[m][r] + bv) * scale;
      int gm = m * 16 + r + 8 * lhalf;
      if (out_mode == 0) {
        int b = gm >> 4, t = gm & 15;
        out[(((b * 16 + panel) * 16 + t) * 64) + wave * 16 + lmod] = y;
      } else {
        out[gm * 1024 + gn] = y;
      }
    }
  }
}

// ---------------------------------------------------------------------------
// Flash attention: one block per (b,h). 4 waves, each owns 32 keys of S and
// a 16-col tile of O. TDM double-buffers K/V chunks of 128 keys.
// ---------------------------------------------------------------------------
__global__ void __launch_bounds__(128)
attn_kernel(const float* __restrict__ qws, const float* __restrict__ pastK,
            const float* __restrict__ pastV, const float* __restrict__ kNew,
            const float* __restrict__ vNew, float* __restrict__ oWs) {
  __shared__ __align__(16) float Kb[2][CHUNK * LDSTR];
  __shared__ __align__(16) float Vb[2][CHUNK * LDSTR];
  __shared__ __align__(16) float Qs[16 * LDSTR];
  __shared__ __align__(16) float Pb[16 * PSTR];
  __shared__ float m_run[16], l_run[16], rowmax_s[16], rowscale[16];
  __shared__ float pmax[128], psum[128];

  const int tid   = threadIdx.x;
  const int lane  = tid & 31;
  const int wave  = tid >> 5;
  const int lhalf = lane >> 4;
  const int lmod  = lane & 15;
  const int bh    = blockIdx.x;          // b*16 + h

  const float* kbase0 = pastK + (size_t)bh * PASTN * HD;
  const float* vbase0 = pastV + (size_t)bh * PASTN * HD;
  const float* kbase1 = kNew + (size_t)bh * 16 * HD;
  const float* vbase1 = vNew + (size_t)bh * 16 * HD;

  auto issue_chunk = [&](int c, int bufsel) {
    const float *kp, *vp;
    u32 rows;
    if (c < 64) { kp = kbase0 + (size_t)c * CHUNK * HD;
                  vp = vbase0 + (size_t)c * CHUNK * HD;
                  rows = (u32)(PASTN - c * CHUNK); }
    else        { kp = kbase1; vp = vbase1; rows = 16u; }
    tdm_load_tile((u32)(uintptr_t)&Kb[bufsel][0], kp, rows);
    tdm_load_tile((u32)(uintptr_t)&Vb[bufsel][0], vp, rows);
  };

  if (wave == 0) issue_chunk(0, 0);

  // q tile (pre-scaled by 1/sqrt(HD) in proj) -> LDS
#pragma unroll
  for (int i = 0; i < 2; ++i) {
    int idx = tid + i * 128, row = idx >> 4, c4 = idx & 15;
    float4 t4 = *(const float4*)(qws + (size_t)bh * 1024 + row * 64 + c4 * 4);
    *(float4*)&Qs[row * LDSTR + c4 * 4] = t4;
  }
  if (tid < 16) { m_run[tid] = -3.0e38f; l_run[tid] = 0.0f; }
  __syncthreads();

  // preload all 16 q A-fragments (row = lmod, K = 4k + 2*lhalf + {0,1})
  v2f aq[16];
#pragma unroll
  for (int k = 0; k < 16; ++k)
    aq[k] = *(const v2f*)&Qs[lmod * LDSTR + 4 * k + 2 * lhalf];

  v8f oacc = {};

  for (int c = 0; c < NCHUNK; ++c) {
    const int buf = c & 1;
    if (wave == 0) {
      if (c + 1 < NCHUNK) {
        issue_chunk(c + 1, buf ^ 1);
        __builtin_amdgcn_s_wait_tensorcnt(2);   // chunk c's 2 TDM ops done
      } else {
        __builtin_amdgcn_s_wait_tensorcnt(0);
      }
    }
    __syncthreads();                            // K/V chunk c visible

    // ---- phase 1: S = q @ K^T (each wave: 2 tiles of 16 keys) ----
#pragma unroll
    for (int t = 0; t < 2; ++t) {
      v8f s = {};
      const int keyrow = wave * 32 + t * 16 + lmod;
#pragma unroll
      for (int k = 0; k < 16; ++k) {
        v2f bk = *(const v2f*)&Kb[buf][keyrow * LDSTR + 4 * k + 2 * lhalf];
        s = __builtin_amdgcn_wmma_f32_16x16x4_f32(
            false, aq[k], false, bk, (short)0, s, false, false);
      }
      if (c == NCHUNK - 1) {
        int keyg = c * CHUNK + keyrow;
        if (keyg >= TOTALK) {
#pragma unroll
          for (int r = 0; r < 8; ++r) s[r] = -3.0e38f;
        }
      }
#pragma unroll
      for (int r = 0; r < 8; ++r)
        Pb[(r + 8 * lhalf) * PSTR + keyrow] = s[r];
    }
    __syncthreads();

    // ---- phase 2: online softmax stats ----
    const int srow = tid & 15, sgrp = tid >> 4;
    {
      float cm = -3.0e38f;
#pragma unroll
      for (int cc = 0; cc < 16; ++cc)
        cm = fmaxf(cm, Pb[srow * PSTR + sgrp * 16 + cc]);
      pmax[sgrp * 16 + srow] = cm;
    }
    __syncthreads();
    if (tid < 16) {
      float cmax = pmax[tid];
#pragma unroll
      for (int g = 1; g < 8; ++g) cmax = fmaxf(cmax, pmax[g * 16 + tid]);
      float m_old = m_run[tid];
      float m_new = fmaxf(m_old, cmax);
      rowmax_s[tid] = m_new;
      rowscale[tid] = __expf(m_old - m_new);
      m_run[tid]    = m_new;
    }
    __syncthreads();
    {
      float mrow = rowmax_s[srow];
      float cs = 0.0f;
#pragma unroll
      for (int cc = 0; cc < 16; ++cc) {
        float p = __expf(Pb[srow * PSTR + sgrp * 16 + cc] - mrow);
        Pb[srow * PSTR + sgrp * 16 + cc] = p;
        cs += p;
      }
      psum[sgrp * 16 + srow] = cs;
    }
    __syncthreads();
    if (tid < 16) {
      float s = 0.0f;
#pragma unroll
      for (int g = 0; g < 8; ++g) s += psum[g * 16 + tid];
      l_run[tid] = l_run[tid] * rowscale[tid] + s;
    }

    // ---- phase 3: O = O*diag(rowscale) + P @ V (wave owns 16 cols of HD) ----
#pragma unroll
    for (int r = 0; r < 8; ++r) oacc[r] *= rowscale[r + 8 * lhalf];
#pragma unroll 8
    for (int k = 0; k < 32; ++k) {
      v2f ap = *(const v2f*)&Pb[lmod * PSTR + 4 * k + 2 * lhalf];
      v2f bv;
      bv.x = Vb[buf][(4 * k + 2 * lhalf) * LDSTR + wave * 16 + lmod];
      bv.y = Vb[buf][(4 * k + 2 * lhalf + 1) * LDSTR + wave * 16 + lmod];
      oacc = __builtin_amdgcn_wmma_f32_16x16x4_f32(
          false, ap, false, bv, (short)0, oacc, false, false);
    }
    __syncthreads();                            // P + buf safe to reuse
  }

  // epilogue: normalize, write [B][T][H*64+hd] (== pre-out-proj row-major)
  const int b = bh >> 4, h = bh & 15;
#pragma unroll
  for (int r = 0; r < 8; ++r) {
    int trow = r + 8 * lhalf;
    float y = oacc[r] / l_run[trow];
    oWs[(size_t)(b * 16 + trow) * 1024 + h * 64 + wave * 16 + lmod] = y;
  }
}

// ---------------------------------------------------------------------------
extern "C" void kernel_launch(void* const* d_in, const int* in_sizes, int n_in,
                              void* d_out, int out_size, void* d_ws, size_t ws_size,
                              hipStream_t stream) {
  (void)in_sizes; (void)n_in; (void)out_size; (void)ws_size;
  const float* x      = (const float*)d_in[0];
  const float* past_k = (const float*)d_in[1];
  const float* past_v = (const float*)d_in[2];
  const float* Wq = (const float*)d_in[3];  const float* bq = (const float*)d_in[4];
  const float* Wk = (const float*)d_in[5];  const float* bk = (const float*)d_in[6];
  const float* Wv = (const float*)d_in[7];  const float* bv = (const float*)d_in[8];
  const float* Wo = (const float*)d_in[9];  const float* bo = (const float*)d_in[10];

  float* out   = (float*)d_out;            // [8,16,1024]
  float* k_out = out + 131072;             // [8,16,16,64]
  float* v_out = out + 262144;             // [8,16,16,64]
  float* q_ws  = (float*)d_ws;             // [8,16,16,64] (pre-scaled q)
  float* a_ws  = q_ws + 131072;            // [128,1024] attention result

  const float inv_sqrt_hd = 0.125f;        // 1/sqrt(64)

  proj_kernel<<<dim3(16), dim3(128), 0, stream>>>(x, Wq, bq, q_ws, 0, inv_sqrt_hd);
  proj_kernel<<<dim3(16), dim3(128), 0, stream>>>(x, Wk, bk, k_out, 0, 1.0f);
  proj_kernel<<<dim3(16), dim3(128), 0, stream>>>(x, Wv, bv, v_out, 0, 1.0f);
  attn_kernel<<<dim3(128), dim3(128), 0, stream>>>(q_ws, past_k, past_v,
                                                   k_out, v_out, a_ws);
  proj_kernel<<<dim3(16), dim3(128), 0, stream>>>(a_ws, Wo, bo, out, 1, 1.0f);
}